// NuanceWeightedRetrieval_37538014167096
// MI455X (gfx1250) — compile-verified
//
#include <hip/hip_runtime.h>
#include <hip/hip_bf16.h>

typedef float v2f __attribute__((ext_vector_type(2)));
typedef float v8f __attribute__((ext_vector_type(8)));

#define D_DIM 768
#define B_DIM 256
#define H_DIM 128
#define TOPK  8
#define NEG_INF (-__builtin_inff())

#define K2_NB     512     // blocks for sim kernel
#define K2_NTILE  64      // corpus rows per tile
#define K2_KC     32      // K chunk
#define LSTRIDE   36      // padded LDS row stride (floats): 144B = 16B-aligned rows,
                          // m*36 mod 64 distinct for m=0..15 -> conflict-free fragments

// ---------------------------------------------------------------------------
// CDNA5 async copy: global -> LDS without VGPR round-trip (ASYNCcnt tracked)
// ---------------------------------------------------------------------------
__device__ __forceinline__ void async_load_b128(void* lds_ptr, const void* gptr) {
  uint32_t lds_off = (uint32_t)(uintptr_t)lds_ptr;   // low 32 bits = LDS byte addr
  uint64_t gaddr = (uint64_t)(uintptr_t)gptr;
  asm volatile("global_load_async_to_lds_b128 %0, %1, off"
               :
               : "v"(lds_off), "v"(gaddr)
               : "memory");
}
__device__ __forceinline__ void wait_async0() {
  asm volatile("s_wait_asynccnt 0" ::: "memory");
}

// ---------------------------------------------------------------------------
// sorted-descending top-8 insertion (register resident)
// ---------------------------------------------------------------------------
__device__ __forceinline__ void topk_insert(float (&sc)[TOPK], int (&ix)[TOPK],
                                            float v, int id) {
  if (v <= sc[TOPK - 1]) return;
  sc[TOPK - 1] = v; ix[TOPK - 1] = id;
#pragma unroll
  for (int t = TOPK - 1; t > 0; --t) {
    if (sc[t] > sc[t - 1]) {
      float tv = sc[t]; sc[t] = sc[t - 1]; sc[t - 1] = tv;
      int   ti = ix[t]; ix[t] = ix[t - 1]; ix[t - 1] = ti;
    }
  }
}

// ---------------------------------------------------------------------------
// K0: normalize queries: qn = q / max(||q||, eps)
// ---------------------------------------------------------------------------
__global__ __launch_bounds__(256) void qnorm_kernel(const float* __restrict__ q,
                                                    float* __restrict__ qn) {
  __shared__ float red[256];
  const int b = blockIdx.x, tid = threadIdx.x;
  float s = 0.f;
  for (int i = tid; i < D_DIM; i += 256) { float x = q[b * D_DIM + i]; s += x * x; }
  red[tid] = s;
  __syncthreads();
  for (int off = 128; off > 0; off >>= 1) {
    if (tid < off) red[tid] += red[tid + off];
    __syncthreads();
  }
  const float inv = 1.f / fmaxf(sqrtf(red[0]), 1e-12f);
  for (int i = tid; i < D_DIM; i += 256) qn[b * D_DIM + i] = q[b * D_DIM + i] * inv;
}

// ---------------------------------------------------------------------------
// K1: per corpus row n: f[n] = sigmoid(relu(c@W1^T+b1)@w2 + b2) / max(||c||,eps)
// 64-row tile, h = C(64x768) x W1^T(768x128) via V_WMMA_F32_16X16X4_F32.
// 8 waves: wave w -> m_sub = w&3, n-group = w>>2 (4 n-subtiles each).
// Staging via global_load_async_to_lds_b128.
// ---------------------------------------------------------------------------
__global__ __launch_bounds__(256) void nuance_kernel(
    const float* __restrict__ corpus, const float* __restrict__ w1,
    const float* __restrict__ b1, const float* __restrict__ w2,
    const float* __restrict__ b2, float* __restrict__ f, int N) {
  __shared__ float cs[64 * LSTRIDE];     // corpus chunk  [64][32] (padded)
  __shared__ float w1s[128 * LSTRIDE];   // W1 chunk      [128][32] (padded)
  __shared__ float hpart[64 * 2];
  __shared__ float nq[64 * 4];

  const int tid = threadIdx.x;
  const int wave = tid >> 5, lane = tid & 31;
  const int half = lane >> 4, l16 = lane & 15;
  const int msub = wave & 3, ngrp = wave >> 2;
  const int n0 = blockIdx.x * 64;

  v8f acc[4];
  const v8f vzero = {0.f, 0.f, 0.f, 0.f, 0.f, 0.f, 0.f, 0.f};
#pragma unroll
  for (int s = 0; s < 4; ++s) acc[s] = vzero;

  float nsq = 0.f;
  const int nrow = tid & 63, ncol0 = (tid >> 6) * 8;

  for (int kc = 0; kc < D_DIM; kc += K2_KC) {
    __syncthreads();
    // corpus chunk: 64x32 floats = 512 x B128, 2 per thread (clamped OOB rows)
#pragma unroll
    for (int i = 0; i < 2; ++i) {
      int idx = tid + i * 256, row = idx >> 3, c4 = idx & 7;
      int g = n0 + row; if (g >= N) g = N - 1;
      async_load_b128(&cs[row * LSTRIDE + c4 * 4],
                      corpus + (size_t)g * D_DIM + kc + c4 * 4);
    }
    // W1 chunk: 128x32 floats = 1024 x B128, 4 per thread
#pragma unroll
    for (int i = 0; i < 4; ++i) {
      int idx = tid + i * 256, row = idx >> 3, c4 = idx & 7;
      async_load_b128(&w1s[row * LSTRIDE + c4 * 4],
                      w1 + row * D_DIM + kc + c4 * 4);
    }
    wait_async0();
    __syncthreads();
    // row-norm partial (deterministic owner: thread = (quarter,row))
#pragma unroll
    for (int j = 0; j < 8; ++j) {
      float x = cs[nrow * LSTRIDE + ncol0 + j];
      nsq += x * x;
    }
    // WMMA over this K chunk
    const float* abase = &cs[(msub * 16 + l16) * LSTRIDE];
#pragma unroll
    for (int k = 0; k < K2_KC; k += 4) {
      v2f a = *(const v2f*)(abase + k + 2 * half);
#pragma unroll
      for (int s = 0; s < 4; ++s) {
        const int ns = ngrp * 4 + s;
        v2f bb = *(const v2f*)(&w1s[(ns * 16 + l16) * LSTRIDE + k + 2 * half]);
        acc[s] = __builtin_amdgcn_wmma_f32_16x16x4_f32(
            false, a, false, bb, (short)0, acc[s], false, false);
      }
    }
  }
  // relu(h+b1)*w2, reduce over j: shfl within 16-lane half, then single-writer LDS
  float rowsum[8];
#pragma unroll
  for (int r = 0; r < 8; ++r) rowsum[r] = 0.f;
#pragma unroll
  for (int s = 0; s < 4; ++s) {
    const int j = (ngrp * 4 + s) * 16 + l16;
    const float b1v = b1[j], w2v = w2[j];
#pragma unroll
    for (int r = 0; r < 8; ++r) {
      float h = acc[s][r] + b1v;
      h = h > 0.f ? h : 0.f;
      rowsum[r] += h * w2v;
    }
  }
#pragma unroll
  for (int r = 0; r < 8; ++r) {
    rowsum[r] += __shfl_xor(rowsum[r], 1, 32);
    rowsum[r] += __shfl_xor(rowsum[r], 2, 32);
    rowsum[r] += __shfl_xor(rowsum[r], 4, 32);
    rowsum[r] += __shfl_xor(rowsum[r], 8, 32);
  }
  if (l16 == 0) {
#pragma unroll
    for (int r = 0; r < 8; ++r)
      hpart[(msub * 16 + half * 8 + r) * 2 + ngrp] = rowsum[r];
  }
  nq[nrow * 4 + (tid >> 6)] = nsq;
  __syncthreads();
  if (tid < 64) {
    const int g = n0 + tid;
    if (g < N) {
      float hs = hpart[tid * 2] + hpart[tid * 2 + 1] + b2[0];
      float sgm = 1.f / (1.f + expf(-hs));
      float nrm = fmaxf(sqrtf(nq[tid * 4] + nq[tid * 4 + 1] + nq[tid * 4 + 2] + nq[tid * 4 + 3]), 1e-12f);
      f[g] = sgm / nrm;
    }
  }
}

// ---------------------------------------------------------------------------
// K2: fused similarity GEMM (f32 WMMA) + per-block top-8.
// 16 waves; wave w owns queries [16w,16w+16); tile = 64 corpus rows.
// LDS 64KB: qs[256][32]+cs[64][32] during GEMM, tile[256][64] overlays after.
// Staging via global_load_async_to_lds_b128 (ASYNCcnt).
// ---------------------------------------------------------------------------
__global__ __launch_bounds__(512) void sim_topk_kernel(
    const float* __restrict__ qn, const float* __restrict__ corpus,
    const float* __restrict__ f, float* __restrict__ psc,
    int* __restrict__ pix, int N) {
  __shared__ float smem[16384];             // 64 KB
  float* qs = smem;                          // [256][LSTRIDE] = 9216 floats
  float* cs = smem + 9216;                   // [64][LSTRIDE]  = 2304 floats
  float* tile = smem;                        // [256][64] overlays (GEMM data dead)

  const int tid = threadIdx.x;
  const int wave = tid >> 5, lane = tid & 31;
  const int half = lane >> 4, l16 = lane & 15;

  float sc[TOPK]; int ix[TOPK];
#pragma unroll
  for (int i = 0; i < TOPK; ++i) { sc[i] = NEG_INF; ix[i] = 0; }

  const int nchunk = (N + K2_NTILE - 1) / K2_NTILE;
  const v8f vzero = {0.f, 0.f, 0.f, 0.f, 0.f, 0.f, 0.f, 0.f};

  for (int ci = blockIdx.x; ci < nchunk; ci += K2_NB) {
    const int n0 = ci * K2_NTILE;
    v8f acc[4];
#pragma unroll
    for (int s = 0; s < 4; ++s) acc[s] = vzero;

    for (int kc = 0; kc < D_DIM; kc += K2_KC) {
      __syncthreads();   // protects tile (prev iter) & qs/cs (readers done)
      // query chunk: 256x32 = 2048 x B128, 4 per thread
#pragma unroll
      for (int i = 0; i < 4; ++i) {
        int idx = tid + i * 512, row = idx >> 3, c4 = idx & 7;
        async_load_b128(&qs[row * LSTRIDE + c4 * 4],
                        qn + row * D_DIM + kc + c4 * 4);
      }
      // corpus chunk: 64x32 = 512 x B128, 1 per thread (clamped OOB rows)
      {
        int row = tid >> 3, c4 = tid & 7;
        int g = n0 + row; if (g >= N) g = N - 1;
        async_load_b128(&cs[row * LSTRIDE + c4 * 4],
                        corpus + (size_t)g * D_DIM + kc + c4 * 4);
      }
      wait_async0();
      __syncthreads();
      const float* abase = &qs[(wave * 16 + l16) * LSTRIDE];
#pragma unroll
      for (int k = 0; k < K2_KC; k += 4) {
        v2f a = *(const v2f*)(abase + k + 2 * half);
#pragma unroll
        for (int s = 0; s < 4; ++s) {
          v2f bb = *(const v2f*)(&cs[(s * 16 + l16) * LSTRIDE + k + 2 * half]);
          acc[s] = __builtin_amdgcn_wmma_f32_16x16x4_f32(
              false, a, false, bb, (short)0, acc[s], false, false);
        }
      }
    }
    __syncthreads();   // all waves done reading qs/cs -> safe to overlay tile
    // scale by nuance/||c|| and dump weighted tile to LDS
#pragma unroll
    for (int s = 0; s < 4; ++s) {
      const int nl = s * 16 + l16;
      const int n = n0 + nl;
      const bool ok = (n < N);
      const float fv = ok ? f[n] : 0.f;
#pragma unroll
      for (int r = 0; r < 8; ++r) {
        const int m = wave * 16 + half * 8 + r;
        tile[m * 64 + nl] = ok ? acc[s][r] * fv : NEG_INF;
      }
    }
    __syncthreads();
    // each lane scans 32 candidates of its query into a register top-8
    {
      const int qrow = wave * 16 + (lane >> 1);
      const int cbase = (lane & 1) * 32;
      const float* trow = &tile[qrow * 64 + cbase];
#pragma unroll 8
      for (int j = 0; j < 32; ++j) topk_insert(sc, ix, trow[j], n0 + cbase + j);
    }
  }
  // write this block's partial candidates (16 per query per block)
  {
    const int qrow = wave * 16 + (lane >> 1);
    const size_t base = ((size_t)qrow * K2_NB + blockIdx.x) * 16 + (size_t)(lane & 1) * 8;
#pragma unroll
    for (int i = 0; i < TOPK; ++i) { psc[base + i] = sc[i]; pix[base + i] = ix[i]; }
  }
}

// ---------------------------------------------------------------------------
// K3: final merge (one block per query) + gather retrieved rows
// ---------------------------------------------------------------------------
__global__ __launch_bounds__(256) void final_topk_kernel(
    const float* __restrict__ psc, const int* __restrict__ pix,
    const float* __restrict__ corpus, float* __restrict__ out_ret,
    float* __restrict__ out_sc, int* __restrict__ out_ix) {
  __shared__ float lsc[256 * TOPK];
  __shared__ int   lix[256 * TOPK];
  __shared__ float fsc[TOPK];
  __shared__ int   fix[TOPK];
  const int q = blockIdx.x, tid = threadIdx.x;
  const int ncand = K2_NB * 16;

  float sc[TOPK]; int ix[TOPK];
#pragma unroll
  for (int i = 0; i < TOPK; ++i) { sc[i] = NEG_INF; ix[i] = 0; }
  const size_t base = (size_t)q * ncand;
  for (int i = tid; i < ncand; i += 256) topk_insert(sc, ix, psc[base + i], pix[base + i]);
#pragma unroll
  for (int i = 0; i < TOPK; ++i) { lsc[tid * TOPK + i] = sc[i]; lix[tid * TOPK + i] = ix[i]; }
  __syncthreads();
  if (tid == 0) {
    float s2[TOPK]; int x2[TOPK];
#pragma unroll
    for (int i = 0; i < TOPK; ++i) { s2[i] = NEG_INF; x2[i] = 0; }
    for (int i = 0; i < 256 * TOPK; ++i) topk_insert(s2, x2, lsc[i], lix[i]);
#pragma unroll
    for (int i = 0; i < TOPK; ++i) { fsc[i] = s2[i]; fix[i] = x2[i]; }
  }
  __syncthreads();
  if (tid < TOPK) {
    out_sc[q * TOPK + tid] = fsc[tid];
    out_ix[q * TOPK + tid] = fix[tid];
  }
  for (int kk = 0; kk < TOPK; ++kk) {
    const float* src = corpus + (size_t)fix[kk] * D_DIM;
    float* dst = out_ret + ((size_t)q * TOPK + kk) * D_DIM;
    for (int i = tid; i < D_DIM; i += 256) dst[i] = src[i];
  }
}

// ---------------------------------------------------------------------------
extern "C" void kernel_launch(void* const* d_in, const int* in_sizes, int n_in,
                              void* d_out, int out_size, void* d_ws, size_t ws_size,
                              hipStream_t stream) {
  const float* q      = (const float*)d_in[0];
  const float* corpus = (const float*)d_in[1];
  const float* w1     = (const float*)d_in[2];
  const float* b1     = (const float*)d_in[3];
  const float* w2     = (const float*)d_in[4];
  const float* b2     = (const float*)d_in[5];
  const int N = in_sizes[1] / D_DIM;

  // workspace layout (floats): f[524288] | qn[256*768] | psc[256*512*16] | pix[...]
  float* f   = (float*)d_ws;
  float* qn  = f + (1 << 19);
  float* psc = qn + (size_t)B_DIM * D_DIM;
  int*   pix = (int*)(psc + (size_t)B_DIM * K2_NB * 16);

  // output layout: retrieved[256*8*768] | scores[256*8] | indices[256*8] (int32)
  float* out_ret = (float*)d_out;
  float* out_sc  = out_ret + (size_t)B_DIM * TOPK * D_DIM;
  int*   out_ix  = (int*)(out_sc + (size_t)B_DIM * TOPK);

  qnorm_kernel<<<B_DIM, 256, 0, stream>>>(q, qn);
  nuance_kernel<<<(N + 63) / 64, 256, 0, stream>>>(corpus, w1, b1, w2, b2, f, N);
  sim_topk_kernel<<<K2_NB, 512, 0, stream>>>(qn, corpus, f, psc, pix, N);
  final_topk_kernel<<<B_DIM, 256, 0, stream>>>(psc, pix, corpus, out_ret, out_sc, out_ix);
}